// NodeSelfAttention_24979529793592
// MI455X (gfx1250) — compile-verified
//
#include <hip/hip_runtime.h>

typedef __attribute__((ext_vector_type(2))) float v2f;
typedef __attribute__((ext_vector_type(8))) float v8f;

#define NN     50000
#define DIMV   256
#define HH     8
#define DH     32
#define EE     800000
#define QKVC   768
#define MBLK   80          // 50000 = 625 * 80; 5 x 16 sub-tiles per wave
#define MSUB   5
#define LSTR   260         // LDS row stride: 1040B = 65*16 (b128-aligned), banks 4l mod 64
#define SCALE  0.17677669529663687f   // 32^-0.5

// ---- monotone float<->uint encoding so atomicMax(uint) orders floats ----
__device__ __forceinline__ unsigned fenc(float f) {
  unsigned u = __float_as_uint(f);
  return (u & 0x80000000u) ? ~u : (u | 0x80000000u);
}
__device__ __forceinline__ float fdec(unsigned u) {
  u = (u & 0x80000000u) ? (u & 0x7FFFFFFFu) : ~u;
  return __uint_as_float(u);
}

// ---------------- Kernel 1: qkv = h @ W_qkv + b_qkv (WMMA f32 16x16x4) ----
// grid: 625 mBlocks * 6 nGroups; block = 256 threads = 8 waves.
// Each wave: one 16-col N tile x five 16-row M sub-tiles (B frag reused 5x).
__global__ __launch_bounds__(256) void k_qkv(
    const float* __restrict__ h, const float* __restrict__ W,
    const float* __restrict__ bias, float* __restrict__ qkv)
{
  extern __shared__ float smem[];          // MBLK x LSTR
  const int mTile  = blockIdx.x / 6;
  const int nGroup = blockIdx.x % 6;
  const int mBase  = mTile * MBLK;
  const int tid    = threadIdx.x;

  // A-tile fill: b128 global loads -> b128 LDS stores.
  for (int i = tid; i < MBLK * (DIMV / 4); i += 256) {
    int r = i >> 6, c = (i & 63) * 4;
    float4 t = *reinterpret_cast<const float4*>(
        h + (size_t)(mBase + r) * DIMV + c);
    *reinterpret_cast<float4*>(&smem[r * LSTR + c]) = t;
  }
  __syncthreads();

  const int wave  = tid >> 5;
  const int lane  = tid & 31;
  const int hs    = lane >> 4;             // half-wave: K pair 0/1 vs 2/3
  const int l     = lane & 15;
  const int nBase = (nGroup * 8 + wave) * 16;

  v8f acc[MSUB] = {};
  #pragma unroll 4
  for (int k = 0; k < DIMV; k += 4) {
    v2f b;
    const float* wp = W + (size_t)(k + 2 * hs) * QKVC + nBase + l;
    b.x = wp[0];
    b.y = wp[QKVC];
    #pragma unroll
    for (int m = 0; m < MSUB; ++m) {
      v2f a;
      const float* ap = &smem[(m * 16 + l) * LSTR + k + 2 * hs];
      a.x = ap[0];
      a.y = ap[1];
      acc[m] = __builtin_amdgcn_wmma_f32_16x16x4_f32(
          false, a, false, b, (short)0, acc[m], false, false);
    }
  }

  const int col = nBase + l;
  const float bv = bias[col];
  #pragma unroll
  for (int m = 0; m < MSUB; ++m) {
    #pragma unroll
    for (int j = 0; j < 8; ++j) {
      int row = mBase + m * 16 + j + 8 * hs;
      qkv[(size_t)row * QKVC + col] = acc[m][j] + bv;
    }
  }
}

// ---------------- Kernel 2: zero accumulators --------------------------
__global__ __launch_bounds__(256) void k_init(
    float* __restrict__ hnew, float* __restrict__ denom,
    unsigned* __restrict__ mx)
{
  int i = blockIdx.x * 256 + threadIdx.x;
  if (i < NN * DIMV) hnew[i] = 0.f;
  if (i < NN * HH) { denom[i] = 0.f; mx[i] = 0u; }
}

// ---------------- Kernel 3: per-(edge,head) score + segment max --------
__global__ __launch_bounds__(256) void k_score(
    const float* __restrict__ qkv, const int* __restrict__ src,
    const int* __restrict__ dst, float* __restrict__ score,
    unsigned* __restrict__ mx)
{
  int idx = blockIdx.x * 256 + threadIdx.x;
  if (idx >= EE * HH) return;
  int e = idx >> 3, hh = idx & 7;
  int d = dst[e], s = src[e];
  const float4* qp = reinterpret_cast<const float4*>(
      qkv + (size_t)d * QKVC + hh * DH);            // q slice
  const float4* kp = reinterpret_cast<const float4*>(
      qkv + (size_t)s * QKVC + DIMV + hh * DH);     // k slice
  float acc = 0.f;
  #pragma unroll
  for (int i = 0; i < 8; ++i) {
    float4 a = qp[i], b = kp[i];
    acc += a.x * b.x + a.y * b.y + a.z * b.z + a.w * b.w;
  }
  acc *= SCALE;
  score[idx] = acc;
  atomicMax(&mx[d * HH + hh], fenc(acc));
}

// ---------------- Kernel 4: exp, denom, unnormalized message scatter ---
__global__ __launch_bounds__(256) void k_edge(
    const float* __restrict__ qkv, const int* __restrict__ src,
    const int* __restrict__ dst, const float* __restrict__ score,
    const unsigned* __restrict__ mx, float* __restrict__ denom,
    float* __restrict__ hnew)
{
  int idx = blockIdx.x * 256 + threadIdx.x;
  if (idx >= EE * HH) return;
  int e = idx >> 3, hh = idx & 7;
  int d = dst[e], s = src[e];
  float m  = fdec(mx[d * HH + hh]);
  float ex = __expf(score[idx] - m);
  atomicAdd(&denom[d * HH + hh], ex);
  const float4* vp = reinterpret_cast<const float4*>(
      qkv + (size_t)s * QKVC + 2 * DIMV + hh * DH);  // v slice
  float4 v[8];
  #pragma unroll
  for (int i = 0; i < 8; ++i) v[i] = vp[i];          // clause of b128 loads
  float* op = hnew + (size_t)d * DIMV + hh * DH;
  #pragma unroll
  for (int i = 0; i < 8; ++i) {
    atomicAdd(&op[4 * i + 0], ex * v[i].x);
    atomicAdd(&op[4 * i + 1], ex * v[i].y);
    atomicAdd(&op[4 * i + 2], ex * v[i].z);
    atomicAdd(&op[4 * i + 3], ex * v[i].w);
  }
}

// ---------------- Kernel 5: out = (normalize+transpose(hnew) + h) @ W_out
__global__ __launch_bounds__(256) void k_out(
    const float* __restrict__ hnew, const float* __restrict__ denom,
    const float* __restrict__ h, const float* __restrict__ W,
    const float* __restrict__ bias, float* __restrict__ out)
{
  extern __shared__ float smem[];          // MBLK x LSTR
  const int mTile  = blockIdx.x >> 1;
  const int nGroup = blockIdx.x & 1;
  const int mBase  = mTile * MBLK;
  const int tid    = threadIdx.x;

  // Fused fill: softmax normalize + (0,2,1) transpose + residual add.
  // One iteration = (row, dd): 8 output cols c = dd*8 + hh, hh = 0..7.
  for (int i = tid; i < MBLK * DH; i += 256) {
    int r = i >> 5, dd = i & 31;
    int row = mBase + r;
    const float4* dn4 = reinterpret_cast<const float4*>(denom + row * HH);
    float4 d0 = dn4[0], d1 = dn4[1];
    const float4* h4 = reinterpret_cast<const float4*>(
        h + (size_t)row * DIMV + dd * 8);
    float4 h0 = h4[0], h1 = h4[1];
    const float* hp = hnew + (size_t)row * DIMV + dd;   // stride DH over hh
    float4 s0, s1;
    s0.x = (d0.x > 0.f ? hp[0 * DH] / d0.x : 0.f) + h0.x;
    s0.y = (d0.y > 0.f ? hp[1 * DH] / d0.y : 0.f) + h0.y;
    s0.z = (d0.z > 0.f ? hp[2 * DH] / d0.z : 0.f) + h0.z;
    s0.w = (d0.w > 0.f ? hp[3 * DH] / d0.w : 0.f) + h0.w;
    s1.x = (d1.x > 0.f ? hp[4 * DH] / d1.x : 0.f) + h1.x;
    s1.y = (d1.y > 0.f ? hp[5 * DH] / d1.y : 0.f) + h1.y;
    s1.z = (d1.z > 0.f ? hp[6 * DH] / d1.z : 0.f) + h1.z;
    s1.w = (d1.w > 0.f ? hp[7 * DH] / d1.w : 0.f) + h1.w;
    *reinterpret_cast<float4*>(&smem[r * LSTR + dd * 8])     = s0;
    *reinterpret_cast<float4*>(&smem[r * LSTR + dd * 8 + 4]) = s1;
  }
  __syncthreads();

  const int wave  = tid >> 5;
  const int lane  = tid & 31;
  const int hs    = lane >> 4;
  const int l     = lane & 15;
  const int nBase = (nGroup * 8 + wave) * 16;

  v8f acc[MSUB] = {};
  #pragma unroll 4
  for (int k = 0; k < DIMV; k += 4) {
    v2f b;
    const float* wp = W + (size_t)(k + 2 * hs) * DIMV + nBase + l;
    b.x = wp[0];
    b.y = wp[DIMV];
    #pragma unroll
    for (int m = 0; m < MSUB; ++m) {
      v2f a;
      const float* ap = &smem[(m * 16 + l) * LSTR + k + 2 * hs];
      a.x = ap[0];
      a.y = ap[1];
      acc[m] = __builtin_amdgcn_wmma_f32_16x16x4_f32(
          false, a, false, b, (short)0, acc[m], false, false);
    }
  }

  const int col = nBase + l;
  const float bv = bias[col];
  #pragma unroll
  for (int m = 0; m < MSUB; ++m) {
    #pragma unroll
    for (int j = 0; j < 8; ++j) {
      int row = mBase + m * 16 + j + 8 * hs;
      out[(size_t)row * DIMV + col] = acc[m][j] + bv;
    }
  }
}

extern "C" void kernel_launch(void* const* d_in, const int* in_sizes, int n_in,
                              void* d_out, int out_size, void* d_ws, size_t ws_size,
                              hipStream_t stream) {
  const float* h    = (const float*)d_in[0];
  const int*   src  = (const int*)d_in[1];
  const int*   dst  = (const int*)d_in[2];
  const float* Wqkv = (const float*)d_in[3];
  const float* bqkv = (const float*)d_in[4];
  const float* Wout = (const float*)d_in[5];
  const float* bout = (const float*)d_in[6];
  float* out = (float*)d_out;

  // workspace layout (floats): qkv | score | mx | denom | hnew
  float*    ws    = (float*)d_ws;
  float*    qkv   = ws;                                   // NN*QKVC
  float*    score = qkv + (size_t)NN * QKVC;              // EE*HH
  unsigned* mx    = (unsigned*)(score + (size_t)EE * HH); // NN*HH
  float*    denom = (float*)(mx + (size_t)NN * HH);       // NN*HH
  float*    hnew  = denom + (size_t)NN * HH;              // NN*DIMV

  const size_t ldsBytes = (size_t)MBLK * LSTR * sizeof(float);  // 83.2 KB

  k_qkv  <<<625 * 6, 256, ldsBytes, stream>>>(h, Wqkv, bqkv, qkv);
  k_init <<<(NN * DIMV + 255) / 256, 256, 0, stream>>>(hnew, denom, mx);
  k_score<<<(EE * HH + 255) / 256,  256, 0, stream>>>(qkv, src, dst, score, mx);
  k_edge <<<(EE * HH + 255) / 256,  256, 0, stream>>>(qkv, src, dst, score, mx, denom, hnew);
  k_out  <<<625 * 2, 256, ldsBytes, stream>>>(hnew, denom, h, Wout, bout, out);
}